// PrototypeRefinement_29557964931396
// MI455X (gfx1250) — compile-verified
//
#include <hip/hip_runtime.h>
#include <hip/hip_bf16.h>
#include <math.h>

typedef float v2f __attribute__((ext_vector_type(2)));
typedef float v8f __attribute__((ext_vector_type(8)));

static constexpr int CB    = 64;    // batch
static constexpr int CF    = 256;   // FEAT
static constexpr int CNC   = 9;
static constexpr int CNQ   = 72;
static constexpr int CKTOP = 4;
static constexpr int CQ    = 45;    // NC*(KTOP+1)
static constexpr int CHW   = 1024;
static constexpr int CHEADS= 8;
static constexpr int CCH   = 32;    // FEAT/HEADS

// ---------------------------------------------------------------- helpers
__device__ __forceinline__ float wave_sum(float v) {
#pragma unroll
  for (int o = 16; o > 0; o >>= 1) v += __shfl_xor(v, o, 32);
  return v;
}
__device__ __forceinline__ float wave_max(float v) {
#pragma unroll
  for (int o = 16; o > 0; o >>= 1) v = fmaxf(v, __shfl_xor(v, o, 32));
  return v;
}
__device__ __forceinline__ float gelu_f(float x) {
  return 0.5f * x * (1.0f + erff(x * 0.70710678118654752440f));
}

// ---------------------------------------------------------------- WMMA GEMM
// C[M,N] = A[M,K] @ B + bias (+gelu) (+res)
//   BT=true : B is W[N,K] row-major (linear layer, x @ W^T)
//   BT=false: B is B[K,N] row-major (plain matmul, e.g. 1x1 conv weights as A)
// BIAS_MODE: 0 none, 1 per-N column, 2 per-M row.
// Uses V_WMMA_F32_16X16X4_F32 (fp32 A/B/C) -> full reference precision.
#define GKC 64
#define GLD 65

template<bool BT, int BIAS_MODE, bool ACT_GELU, bool HAS_RES>
__global__ __launch_bounds__(256) void gemm_wmma_kernel(
    const float* __restrict__ A, long sA_,
    const float* __restrict__ Bm, long sB_,
    const float* __restrict__ bias,
    const float* __restrict__ res, long sR_,
    float* __restrict__ C, long sC_,
    int M, int N, int K)
{
  __shared__ float sA[64 * GLD];
  __shared__ float sB[32 * GLD];
  const int tid  = threadIdx.x;
  const int lane = tid & 31;
  const int wv   = tid >> 5;
  const int wm   = wv >> 1;       // 0..3
  const int wn   = wv & 1;        // 0..1
  const int m0   = blockIdx.y * 64;
  const int n0   = blockIdx.x * 32;
  const int b    = blockIdx.z;

  const float* Ab = A + (long)b * sA_;
  const float* Bb = Bm + (long)b * sB_;
  float*       Cb = C + (long)b * sC_;

  v8f acc = {0.f, 0.f, 0.f, 0.f, 0.f, 0.f, 0.f, 0.f};

  const int half = lane >> 4;       // 0 or 1
  const int l16  = lane & 15;
  const int mrow = wm * 16 + l16;
  const int ncol = wn * 16 + l16;

  for (int kc = 0; kc < K; kc += GKC) {
    // cooperative load A: 64 rows x 64 k (zero-padded)
#pragma unroll
    for (int p = 0; p < 16; ++p) {
      int idx = tid + p * 256;
      int r = idx >> 6, kk = idx & 63;
      int gr = m0 + r, gk = kc + kk;
      float v = 0.f;
      if (gr < M && gk < K) v = Ab[(long)gr * K + gk];
      sA[r * GLD + kk] = v;
    }
    // cooperative load B into [n][k] layout
    if (BT) {
#pragma unroll
      for (int p = 0; p < 8; ++p) {
        int idx = tid + p * 256;
        int n = idx >> 6, kk = idx & 63;
        int gn = n0 + n, gk = kc + kk;
        float v = 0.f;
        if (gn < N && gk < K) v = Bb[(long)gn * K + gk];
        sB[n * GLD + kk] = v;
      }
    } else {
#pragma unroll
      for (int p = 0; p < 8; ++p) {
        int idx = tid + p * 256;
        int n = idx & 31, kk = idx >> 5;
        int gn = n0 + n, gk = kc + kk;
        float v = 0.f;
        if (gn < N && gk < K) v = Bb[(long)gk * N + gn];
        sB[n * GLD + kk] = v;
      }
    }
    __syncthreads();
    // 16 x V_WMMA_F32_16X16X4_F32 per chunk
#pragma unroll
    for (int k0 = 0; k0 < GKC; k0 += 4) {
      v2f a, bb;
      a.x  = sA[mrow * GLD + k0 + 2 * half];
      a.y  = sA[mrow * GLD + k0 + 2 * half + 1];
      bb.x = sB[ncol * GLD + k0 + 2 * half];
      bb.y = sB[ncol * GLD + k0 + 2 * half + 1];
      acc = __builtin_amdgcn_wmma_f32_16x16x4_f32(
          false, a, false, bb, (short)0, acc, false, false);
    }
    __syncthreads();
  }

  const int gn = n0 + ncol;
#pragma unroll
  for (int r = 0; r < 8; ++r) {
    int gm = m0 + wm * 16 + r + 8 * half;
    if (gm < M && gn < N) {
      float v = acc[r];
      if (BIAS_MODE == 1) v += bias[gn];
      else if (BIAS_MODE == 2) v += bias[gm];
      if (ACT_GELU) v = gelu_f(v);
      if (HAS_RES) v += res[(long)b * sR_ + (long)gm * N + gn];
      Cb[(long)gm * N + gn] = v;
    }
  }
}

// ---------------------------------------------------------------- small kernels
__global__ void add_kernel(const float* __restrict__ a, const float* __restrict__ b,
                           float* __restrict__ c, long n)
{
  long i = (long)blockIdx.x * 256 + threadIdx.x;
  if (i < n) c[i] = a[i] + b[i];
}

// LayerNorm over contiguous last dim of 256
__global__ void rowln_kernel(const float* __restrict__ in, float* __restrict__ out,
                             const float* __restrict__ g, const float* __restrict__ be,
                             int rows)
{
  int gw = (int)(((long)blockIdx.x * blockDim.x + threadIdx.x) >> 5);
  int lane = threadIdx.x & 31;
  if (gw >= rows) return;
  const float* x = in + (long)gw * CF;
  float v[8], s = 0.f, s2 = 0.f;
#pragma unroll
  for (int j = 0; j < 8; ++j) { v[j] = x[lane + 32 * j]; s += v[j]; s2 += v[j] * v[j]; }
  s = wave_sum(s); s2 = wave_sum(s2);
  float mean = s * (1.f / CF);
  float var  = s2 * (1.f / CF) - mean * mean;
  float inv  = rsqrtf(var + 1e-5f);
  float* o = out + (long)gw * CF;
#pragma unroll
  for (int j = 0; j < 8; ++j) {
    int c = lane + 32 * j;
    o[c] = (v[j] - mean) * inv * g[c] + be[c];
  }
}

// L2 norm of rows of length 256
__global__ void rownorm_kernel(const float* __restrict__ in, float* __restrict__ out, int rows)
{
  int gw = (int)(((long)blockIdx.x * blockDim.x + threadIdx.x) >> 5);
  int lane = threadIdx.x & 31;
  if (gw >= rows) return;
  const float* x = in + (long)gw * CF;
  float s = 0.f;
#pragma unroll
  for (int j = 0; j < 8; ++j) { float v = x[lane + 32 * j]; s += v * v; }
  s = wave_sum(s);
  if (lane == 0) out[gw] = sqrtf(s);
}

// LayerNorm over channel axis for [B, 256, HW] (one wave per pixel)
__global__ void chanln_kernel(const float* __restrict__ in, float* __restrict__ out,
                              const float* __restrict__ g, const float* __restrict__ be)
{
  int gw = (int)(((long)blockIdx.x * blockDim.x + threadIdx.x) >> 5);  // 0..B*HW-1
  int lane = threadIdx.x & 31;
  int b = gw >> 10, p = gw & 1023;
  const float* x = in + (long)b * CF * CHW + p;
  float v[8], s = 0.f, s2 = 0.f;
#pragma unroll
  for (int j = 0; j < 8; ++j) {
    v[j] = x[(long)(lane + 32 * j) * CHW];
    s += v[j]; s2 += v[j] * v[j];
  }
  s = wave_sum(s); s2 = wave_sum(s2);
  float mean = s * (1.f / CF);
  float var  = s2 * (1.f / CF) - mean * mean;
  float inv  = rsqrtf(var + 1e-5f);
  float* o = out + (long)b * CF * CHW + p;
#pragma unroll
  for (int j = 0; j < 8; ++j) {
    int c = lane + 32 * j;
    o[(long)c * CHW] = (v[j] - mean) * inv * g[c] + be[c];
  }
}

// cosine similarity + top-4 per (b, class)
__global__ __launch_bounds__(256) void costopk_kernel(
    const float* __restrict__ inter_lin, const float* __restrict__ intra_lin,
    const float* __restrict__ inorm, const float* __restrict__ anorm,
    int* __restrict__ topk)
{
  __shared__ float cos_s[CNQ];
  int bc = blockIdx.x;                 // b*NC + cls
  int b  = bc / CNC;
  int tid = threadIdx.x, lane = tid & 31, wv = tid >> 5;
  const float* ar = intra_lin + (long)bc * CF;
  float an = fmaxf(anorm[bc], 1e-12f);
  for (int q = wv; q < CNQ; q += 8) {
    const float* qr = inter_lin + ((long)b * CNQ + q) * CF;
    float s = 0.f;
    for (int i = lane; i < CF; i += 32) s += ar[i] * qr[i];
    s = wave_sum(s);
    if (lane == 0) cos_s[q] = s / (an * fmaxf(inorm[b * CNQ + q], 1e-12f));
  }
  __syncthreads();
  if (tid == 0) {
    bool tk[CNQ];
    for (int q = 0; q < CNQ; ++q) tk[q] = false;
    for (int k = 0; k < CKTOP; ++k) {
      float best = -3.4e38f; int bi = 0;
      for (int q = 0; q < CNQ; ++q)
        if (!tk[q] && cos_s[q] > best) { best = cos_s[q]; bi = q; }
      tk[bi] = true;
      topk[bc * CKTOP + k] = bi;
    }
  }
}

// protos[b, c*5+s, :] : s==0 -> intra row, else gathered inter row
__global__ void gather_protos_kernel(const float* __restrict__ inter_lin,
                                     const float* __restrict__ intra_lin,
                                     const int* __restrict__ topk,
                                     float* __restrict__ protos)
{
  long i = (long)blockIdx.x * 256 + threadIdx.x;
  if (i >= (long)CB * CQ * CF) return;
  int f = (int)(i & 255);
  long t = i >> 8;
  int qq = (int)(t % CQ);
  int b  = (int)(t / CQ);
  int cls = qq / 5, s = qq % 5;
  float v;
  if (s == 0) v = intra_lin[((long)b * CNC + cls) * CF + f];
  else {
    int j = topk[(b * CNC + cls) * CKTOP + (s - 1)];
    v = inter_lin[((long)b * CNQ + j) * CF + f];
  }
  protos[i] = v;
}

// concat image(256ch) + mask(9ch) -> [B,265,HW]
__global__ void concat_x_kernel(const float* __restrict__ img,
                                const float* __restrict__ mem,
                                float* __restrict__ xcat)
{
  long i = (long)blockIdx.x * 256 + threadIdx.x;
  if (i >= (long)CB * 265 * CHW) return;
  int p = (int)(i % CHW);
  long t = i / CHW;
  int c = (int)(t % 265);
  int b = (int)(t / 265);
  xcat[i] = (c < CF) ? img[((long)b * CF + c) * CHW + p]
                     : mem[((long)b * CNC + (c - CF)) * CHW + p];
}

// depthwise 3x3 SAME
__global__ void dwconv_kernel(const float* __restrict__ in, float* __restrict__ out,
                              const float* __restrict__ w, const float* __restrict__ bias)
{
  long i = (long)blockIdx.x * 256 + threadIdx.x;
  if (i >= (long)CB * CF * CHW) return;
  int x = (int)(i & 31);
  int y = (int)((i >> 5) & 31);
  int c = (int)((i >> 10) & 255);
  int b = (int)(i >> 18);
  const float* base = in + ((long)b * CF + c) * CHW;
  const float* wc = w + c * 9;
  float s = bias[c];
#pragma unroll
  for (int dy = -1; dy <= 1; ++dy) {
    int yy = y + dy;
    if (yy < 0 || yy > 31) continue;
#pragma unroll
    for (int dx = -1; dx <= 1; ++dx) {
      int xx = x + dx;
      if (xx < 0 || xx > 31) continue;
      s += wc[(dy + 1) * 3 + (dx + 1)] * base[yy * 32 + xx];
    }
  }
  out[i] = s;
}

// [B,256,1024] -> [B,1024,256]
__global__ void transpose_kernel(const float* __restrict__ in, float* __restrict__ out)
{
  __shared__ float tile[32][33];
  int b  = blockIdx.z;
  int c0 = blockIdx.y * 32;
  int p0 = blockIdx.x * 32;
  int tx = threadIdx.x & 31, ty = threadIdx.x >> 5;  // ty 0..7
#pragma unroll
  for (int j = 0; j < 32; j += 8)
    tile[ty + j][tx] = in[((long)b * CF + c0 + ty + j) * CHW + p0 + tx];
  __syncthreads();
#pragma unroll
  for (int j = 0; j < 32; j += 8)
    out[((long)b * CHW + p0 + ty + j) * CF + c0 + tx] = tile[tx][ty + j];
}

// per-pixel 1x1 conv reductions: s_img[b,p], s_msk[b,p]
__global__ void attvec_kernel(const float* __restrict__ img, const float* __restrict__ mem,
                              const float* __restrict__ icw, const float* __restrict__ icb,
                              const float* __restrict__ mcw, const float* __restrict__ mcb,
                              float* __restrict__ s_img, float* __restrict__ s_msk)
{
  long i = (long)blockIdx.x * 256 + threadIdx.x;
  if (i >= (long)CB * CHW) return;
  int b = (int)(i >> 10), p = (int)(i & 1023);
  const float* ib = img + (long)b * CF * CHW + p;
  float s = icb[0];
  for (int c = 0; c < CF; ++c) s += icw[c] * ib[(long)c * CHW];
  s_img[i] = s;
  const float* mb = mem + (long)b * CNC * CHW + p;
  float t = mcb[0];
#pragma unroll
  for (int k = 0; k < CNC; ++k) t += mcw[k] * mb[(long)k * CHW];
  s_msk[i] = t;
}

// att_img[b,j] = s_img[b,:] . ip_w[j,:] + ip_b[j]  (and same for mask)
__global__ void attproj_kernel(const float* __restrict__ s_img, const float* __restrict__ s_msk,
                               const float* __restrict__ ipw, const float* __restrict__ ipb,
                               const float* __restrict__ mpw, const float* __restrict__ mpb,
                               float* __restrict__ att_img, float* __restrict__ att_msk)
{
  int gw = (int)(((long)blockIdx.x * blockDim.x + threadIdx.x) >> 5);  // 0..B*32-1
  int lane = threadIdx.x & 31;
  int b = gw >> 5, j = gw & 31;
  float s = 0.f, t = 0.f;
  for (int p = lane; p < CHW; p += 32) {
    s += s_img[(long)b * CHW + p] * ipw[(long)j * CHW + p];
    t += s_msk[(long)b * CHW + p] * mpw[(long)j * CHW + p];
  }
  s = wave_sum(s); t = wave_sum(t);
  if (lane == 0) {
    att_img[b * 32 + j] = s + ipb[j];
    att_msk[b * 32 + j] = t + mpb[j];
  }
}

// full attention for one (b,h): att = softmax((class*alpha + masked qk*beta)/sqrt(32)) @ V
__global__ __launch_bounds__(256) void attention_kernel(
    const float* __restrict__ qbuf, const float* __restrict__ kbuf,
    const float* __restrict__ vbuf, const float* __restrict__ masks,
    const float* __restrict__ att_img, const float* __restrict__ att_msk,
    const float* __restrict__ cc_w, const float* __restrict__ cc_b,
    const float* __restrict__ alpha, const float* __restrict__ beta,
    float* __restrict__ attout)
{
  __shared__ float qs[CQ * CCH];
  __shared__ float att_s[CHW];
  __shared__ float red[256];
  __shared__ float wred[8];
  __shared__ float ai_s[32], am_s[32];
  const int tid = threadIdx.x, lane = tid & 31, wv = tid >> 5;
  const int b = blockIdx.x >> 3, h = blockIdx.x & 7;

  for (int i = tid; i < CQ * CCH; i += 256) {
    int q = i >> 5, c = i & 31;
    qs[i] = qbuf[((long)b * CQ + q) * CF + h * CCH + c];
  }
  if (tid < 32) { ai_s[tid] = att_img[b * 32 + tid]; am_s[tid] = att_msk[b * 32 + tid]; }
  __syncthreads();

  const float invs = 0.17677669529663688f;  // 1/sqrt(32)
  for (int q = 0; q < CQ; ++q) {
    int cls = q / 5;
    float cw = cc_w[q], cb = cc_b[q];
    const float* mrow = masks + ((long)b * CNC + cls) * CHW;
    float lmax = -3.4e38f;
    for (int p = tid; p < CHW; p += 256) {
      float ac = ai_s[p >> 5] * am_s[p & 31];
      const float* kp = kbuf + ((long)b * CHW + p) * CF + h * CCH;
      float dot = 0.f;
#pragma unroll
      for (int c = 0; c < CCH; ++c) dot += qs[q * CCH + c] * kp[c];
      float a = ((cw * ac + cb) * alpha[p] + dot * mrow[p] * beta[p]) * invs;
      att_s[p] = a;
      lmax = fmaxf(lmax, a);
    }
    lmax = wave_max(lmax);
    if (lane == 0) wred[wv] = lmax;
    __syncthreads();
    float bmax = wred[0];
#pragma unroll
    for (int i = 1; i < 8; ++i) bmax = fmaxf(bmax, wred[i]);
    __syncthreads();
    float lsum = 0.f;
    for (int p = tid; p < CHW; p += 256) {
      float e = expf(att_s[p] - bmax);
      att_s[p] = e;
      lsum += e;
    }
    lsum = wave_sum(lsum);
    if (lane == 0) wred[wv] = lsum;
    __syncthreads();
    float bsum = 0.f;
#pragma unroll
    for (int i = 0; i < 8; ++i) bsum += wred[i];
    float inv = 1.f / bsum;

    int c = tid & 31, chunk = tid >> 5;
    float part = 0.f;
    const float* vb = vbuf + (long)b * CHW * CF + h * CCH + c;
    for (int p = chunk * 128; p < chunk * 128 + 128; ++p)
      part += att_s[p] * vb[(long)p * CF];
    red[tid] = part;
    __syncthreads();
    if (tid < 32) {
      float o = 0.f;
#pragma unroll
      for (int j = 0; j < 8; ++j) o += red[j * 32 + tid];
      attout[((long)b * CQ + q) * CF + h * CCH + tid] = o * inv;
    }
    __syncthreads();
  }
}

// cat1 rows 0..71 = inter_lin + scatter-add(protos2 slots 1..4) * inter_scale
__global__ void inter_update_kernel(const float* __restrict__ inter_lin,
                                    const float* __restrict__ protos2,
                                    const int* __restrict__ topk,
                                    const float* __restrict__ pscale,
                                    float* __restrict__ cat1)
{
  int b = blockIdx.x;
  int f = threadIdx.x;  // 256
  float* dst = cat1 + (long)b * 81 * CF;
  const float* src = inter_lin + (long)b * CNQ * CF;
  for (int q = 0; q < CNQ; ++q) dst[q * CF + f] = src[q * CF + f];
  float sc = pscale[f];
  for (int cls = 0; cls < CNC; ++cls)
#pragma unroll
    for (int s = 0; s < CKTOP; ++s) {
      int idx = topk[(b * CNC + cls) * CKTOP + s];
      dst[idx * CF + f] += sc * protos2[((long)b * CQ + cls * 5 + 1 + s) * CF + f];
    }
}

// cat1 rows 72..80 = mask_inter[b,n,c] = sum_p image[b,c,p]*masks[b,n,p]
__global__ __launch_bounds__(256) void mask_inter_kernel(const float* __restrict__ img,
                                                         const float* __restrict__ masks,
                                                         float* __restrict__ cat1)
{
  __shared__ float mrow[CHW];
  int bn = blockIdx.x;
  int b = bn / CNC, n = bn % CNC;
  int tid = threadIdx.x, lane = tid & 31, wv = tid >> 5;
  for (int p = tid; p < CHW; p += 256) mrow[p] = masks[((long)b * CNC + n) * CHW + p];
  __syncthreads();
  for (int cc = 0; cc < 32; ++cc) {
    int c = wv * 32 + cc;
    const float* ib = img + ((long)b * CF + c) * CHW;
    float s = 0.f;
    for (int p = lane; p < CHW; p += 32) s += ib[p] * mrow[p];
    s = wave_sum(s);
    if (lane == 0) cat1[(long)b * 81 * CF + (72 + n) * CF + c] = s;
  }
}

// cat2[b,n,0:256] = intra_lin + protos2[:,n*5,:]*intra_scale ; cat2[b,n,256+k] = mask_intra
__global__ __launch_bounds__(256) void intra_cat_kernel(const float* __restrict__ mem,
                                                        const float* __restrict__ masks,
                                                        const float* __restrict__ intra_lin,
                                                        const float* __restrict__ protos2,
                                                        const float* __restrict__ pscale,
                                                        float* __restrict__ cat2)
{
  __shared__ float mrow[CHW];
  int bn = blockIdx.x;
  int b = bn / CNC, n = bn % CNC;
  int tid = threadIdx.x, lane = tid & 31, wv = tid >> 5;
  for (int p = tid; p < CHW; p += 256) mrow[p] = masks[((long)b * CNC + n) * CHW + p];
  __syncthreads();
  {
    int f = tid;
    float v = intra_lin[((long)b * CNC + n) * CF + f] +
              protos2[((long)b * CQ + n * 5) * CF + f] * pscale[f];
    cat2[(long)bn * 265 + f] = v;
  }
  for (int k = wv; k < CNC; k += 8) {
    const float* mb = mem + ((long)b * CNC + k) * CHW;
    float s = 0.f;
    for (int p = lane; p < CHW; p += 32) s += mb[p] * mrow[p];
    s = wave_sum(s);
    if (lane == 0) cat2[(long)bn * 265 + CF + k] = s;
  }
}

// ---------------------------------------------------------------- host side
template<bool BT, int BIAS, bool GE, bool RES>
static inline void launch_gemm(hipStream_t st,
                               const float* A_, long sA_, const float* B_, long sB_,
                               const float* bias, const float* res, long sR_,
                               float* C_, long sC_, int M, int N, int K, int batch)
{
  dim3 g(N / 32, (M + 63) / 64, batch);
  gemm_wmma_kernel<BT, BIAS, GE, RES><<<g, dim3(256), 0, st>>>(
      A_, sA_, B_, sB_, bias, res, sR_, C_, sC_, M, N, K);
}

extern "C" void kernel_launch(void* const* d_in, const int* in_sizes, int n_in,
                              void* d_out, int out_size, void* d_ws, size_t ws_size,
                              hipStream_t stream)
{
  (void)in_sizes; (void)n_in; (void)out_size; (void)ws_size;

  // tensor inputs
  const float* image  = (const float*)d_in[0];
  const float* maskem = (const float*)d_in[1];
  const float* interP = (const float*)d_in[2];
  const float* intraP = (const float*)d_in[3];
  const float* interE = (const float*)d_in[4];
  const float* intraE = (const float*)d_in[5];
  const float* masks  = (const float*)d_in[6];
  // params (dict order)
  const float* in1_w = (const float*)d_in[7];
  const float* in1_b = (const float*)d_in[8];
  const float* in2_w = (const float*)d_in[9];
  const float* in2_b = (const float*)d_in[10];
  const float* qn_g  = (const float*)d_in[11];
  const float* qn_b  = (const float*)d_in[12];
  const float* q_w   = (const float*)d_in[13];
  const float* q_b   = (const float*)d_in[14];
  const float* xp1_w = (const float*)d_in[15];
  const float* xp1_b = (const float*)d_in[16];
  const float* xp2_w = (const float*)d_in[17];
  const float* xp2_b = (const float*)d_in[18];
  const float* xp2_g = (const float*)d_in[19];
  const float* xp2_b2= (const float*)d_in[20];
  const float* xc_w  = (const float*)d_in[21];
  const float* xc_b  = (const float*)d_in[22];
  const float* k_w   = (const float*)d_in[23];
  const float* k_b   = (const float*)d_in[24];
  const float* v_w   = (const float*)d_in[25];
  const float* v_b   = (const float*)d_in[26];
  const float* ic_w  = (const float*)d_in[27];
  const float* ic_b  = (const float*)d_in[28];
  const float* mc_w  = (const float*)d_in[29];
  const float* mc_b  = (const float*)d_in[30];
  const float* ip_w  = (const float*)d_in[31];
  const float* ip_b  = (const float*)d_in[32];
  const float* mp_w  = (const float*)d_in[33];
  const float* mp_b  = (const float*)d_in[34];
  const float* cc_w  = (const float*)d_in[35];
  const float* cc_b  = (const float*)d_in[36];
  const float* alpha = (const float*)d_in[37];
  const float* beta  = (const float*)d_in[38];
  const float* op_w  = (const float*)d_in[39];
  const float* op_b  = (const float*)d_in[40];
  const float* p_inter = (const float*)d_in[41];
  const float* p_intra = (const float*)d_in[42];
  const float* l1_w  = (const float*)d_in[43];
  const float* l1_b  = (const float*)d_in[44];
  const float* l2_w  = (const float*)d_in[45];
  const float* l2_b  = (const float*)d_in[46];
  const float* f1_g  = (const float*)d_in[47];
  const float* f1_b  = (const float*)d_in[48];
  const float* f1_w1 = (const float*)d_in[49];
  const float* f1_b1 = (const float*)d_in[50];
  const float* f1_w2 = (const float*)d_in[51];
  const float* f1_b2 = (const float*)d_in[52];
  const float* f2_g  = (const float*)d_in[53];
  const float* f2_b  = (const float*)d_in[54];
  const float* f2_w1 = (const float*)d_in[55];
  const float* f2_b1 = (const float*)d_in[56];
  const float* f2_w2 = (const float*)d_in[57];
  const float* f2_b2 = (const float*)d_in[58];

  float* out_inter = (float*)d_out;                         // [64,72,256]
  float* out_intra = (float*)d_out + (long)CB * CNQ * CF;   // [64,9,256]

  // ------- workspace layout (floats), with reuse aliasing -------
  float* ws = (float*)d_ws;
  size_t off = 0;
  auto alloc = [&](size_t n) -> float* {
    float* p = ws + off;
    off += (n + 255) & ~(size_t)255;
    return p;
  };
  float* tmpA      = alloc((size_t)CB * CNQ * CF);   // also reused as lnbuf
  float* inter_lin = alloc((size_t)CB * CNQ * CF);
  float* tmpB      = alloc((size_t)CB * CNC * 265);  // intra add, later cat2
  float* intra_lin = alloc((size_t)CB * CNC * CF);
  float* inorm     = alloc((size_t)CB * CNQ);
  float* anorm     = alloc((size_t)CB * CNC);
  int*   topk      = (int*)alloc((size_t)CB * CNC * CKTOP);
  float* protos    = alloc((size_t)CB * CQ * CF);    // later reused as protos2
  float* protosln  = alloc((size_t)CB * CQ * CF);    // later reused as attout
  float* qbuf      = alloc((size_t)CB * CQ * CF);
  float* s_img     = alloc((size_t)CB * CHW);
  float* s_msk     = alloc((size_t)CB * CHW);
  float* att_img   = alloc((size_t)CB * 32);
  float* att_msk   = alloc((size_t)CB * 32);
  float* cat1      = alloc((size_t)CB * 81 * CF);
  float* inter3    = alloc((size_t)CB * CNQ * CF);
  float* intra3    = alloc((size_t)CB * CNC * CF);
  float* bufA      = alloc((size_t)CB * 265 * CHW);  // xcat -> xp2 out -> dwconv out
  float* bufB      = alloc((size_t)CB * CF * CHW);   // xp1 out -> ln out -> xT -> h1
  float* bufC      = alloc((size_t)CB * CHW * CF);   // kbuf
  float* bufD      = alloc((size_t)CB * CHW * CF);   // vbuf
  float* lnbuf   = tmpA;      // [<=64*72*256]
  float* cat2    = tmpB;      // [64,9,265]
  float* protos2 = protos;    // [64,45,256]
  float* attout  = protosln;  // [64,45,256]
  float* h1      = bufB;      // [<=64*72*1024]

  dim3 blk(256);

  // ---- stage 1: input linears (WMMA, fp32) ----
  {
    long n = (long)CB * CNQ * CF;
    add_kernel<<<dim3((unsigned)((n + 255) / 256)), blk, 0, stream>>>(interP, interE, tmpA, n);
    launch_gemm<true, 1, false, false>(stream, tmpA, 0, in1_w, 0, in1_b, nullptr, 0,
                                       inter_lin, 0, CB * CNQ, CF, CF, 1);
    long m = (long)CB * CNC * CF;
    add_kernel<<<dim3((unsigned)((m + 255) / 256)), blk, 0, stream>>>(intraP, intraE, tmpB, m);
    launch_gemm<true, 1, false, false>(stream, tmpB, 0, in2_w, 0, in2_b, nullptr, 0,
                                       intra_lin, 0, CB * CNC, CF, CF, 1);
  }

  // ---- stage 2: cosine + top-k ----
  rownorm_kernel<<<dim3((CB * CNQ + 7) / 8), blk, 0, stream>>>(inter_lin, inorm, CB * CNQ);
  rownorm_kernel<<<dim3((CB * CNC + 7) / 8), blk, 0, stream>>>(intra_lin, anorm, CB * CNC);
  costopk_kernel<<<dim3(CB * CNC), blk, 0, stream>>>(inter_lin, intra_lin, inorm, anorm, topk);

  // ---- stage 3: protos gather, LN, q projection ----
  {
    long n = (long)CB * CQ * CF;
    gather_protos_kernel<<<dim3((unsigned)((n + 255) / 256)), blk, 0, stream>>>(
        inter_lin, intra_lin, topk, protos);
  }
  rowln_kernel<<<dim3((CB * CQ + 7) / 8), blk, 0, stream>>>(protos, protosln, qn_g, qn_b, CB * CQ);
  launch_gemm<true, 1, false, false>(stream, protosln, 0, q_w, 0, q_b, nullptr, 0,
                                     qbuf, 0, CB * CQ, CF, CF, 1);

  // ---- stage 4: pixel pipeline ----
  {
    long n = (long)CB * 265 * CHW;
    concat_x_kernel<<<dim3((unsigned)((n + 255) / 256)), blk, 0, stream>>>(image, maskem, bufA);
  }
  // xp1: per-batch NN GEMM, A = weights [256,265], B = xcat[b] [265,1024], gelu
  launch_gemm<false, 2, true, false>(stream, xp1_w, 0, bufA, (long)265 * CHW,
                                     xp1_b, nullptr, 0, bufB, (long)CF * CHW,
                                     CF, CHW, 265, CB);
  // xp2: NN GEMM 256->256
  launch_gemm<false, 2, false, false>(stream, xp2_w, 0, bufB, (long)CF * CHW,
                                      xp2_b, nullptr, 0, bufA, (long)CF * CHW,
                                      CF, CHW, CF, CB);
  chanln_kernel<<<dim3(CB * CHW / 8), blk, 0, stream>>>(bufA, bufB, xp2_g, xp2_b2);
  {
    long n = (long)CB * CF * CHW;
    dwconv_kernel<<<dim3((unsigned)((n + 255) / 256)), blk, 0, stream>>>(bufB, bufA, xc_w, xc_b);
  }
  transpose_kernel<<<dim3(CHW / 32, CF / 32, CB), blk, 0, stream>>>(bufA, bufB);
  // k/v projections (batch folded into M)
  launch_gemm<true, 1, false, false>(stream, bufB, 0, k_w, 0, k_b, nullptr, 0,
                                     bufC, 0, CB * CHW, CF, CF, 1);
  launch_gemm<true, 1, false, false>(stream, bufB, 0, v_w, 0, v_b, nullptr, 0,
                                     bufD, 0, CB * CHW, CF, CF, 1);

  // ---- stage 5: class-attention scalars ----
  {
    long n = (long)CB * CHW;
    attvec_kernel<<<dim3((unsigned)((n + 255) / 256)), blk, 0, stream>>>(
        image, maskem, ic_w, ic_b, mc_w, mc_b, s_img, s_msk);
  }
  attproj_kernel<<<dim3(CB * 32 / 8), blk, 0, stream>>>(s_img, s_msk, ip_w, ip_b,
                                                        mp_w, mp_b, att_img, att_msk);

  // ---- stage 6: attention + output projection ----
  attention_kernel<<<dim3(CB * CHEADS), blk, 0, stream>>>(
      qbuf, bufC, bufD, masks, att_img, att_msk, cc_w, cc_b, alpha, beta, attout);
  launch_gemm<true, 1, false, false>(stream, attout, 0, op_w, 0, op_b, nullptr, 0,
                                     protos2, 0, CB * CQ, CF, CF, 1);

  // ---- stage 7: prototype updates + mixing linears ----
  inter_update_kernel<<<dim3(CB), blk, 0, stream>>>(inter_lin, protos2, topk, p_inter, cat1);
  mask_inter_kernel<<<dim3(CB * CNC), blk, 0, stream>>>(image, masks, cat1);
  // l1: per-batch NN GEMM, A = l1_w [72,81], B = cat1[b] [81,256], bias per-row(q)
  launch_gemm<false, 2, false, false>(stream, l1_w, 0, cat1, (long)81 * CF,
                                      l1_b, nullptr, 0, inter3, (long)CNQ * CF,
                                      CNQ, CF, 81, CB);
  intra_cat_kernel<<<dim3(CB * CNC), blk, 0, stream>>>(maskem, masks, intra_lin,
                                                       protos2, p_intra, cat2);
  // l2: NT GEMM, K=265 (zero-padded chunks)
  launch_gemm<true, 1, false, false>(stream, cat2, 0, l2_w, 0, l2_b, nullptr, 0,
                                     intra3, 0, CB * CNC, CF, 265, 1);

  // ---- stage 8: FFNs (fused GELU + fused residual into d_out) ----
  rowln_kernel<<<dim3((CB * CNQ + 7) / 8), blk, 0, stream>>>(inter3, lnbuf, f1_g, f1_b, CB * CNQ);
  launch_gemm<true, 1, true, false>(stream, lnbuf, 0, f1_w1, 0, f1_b1, nullptr, 0,
                                    h1, 0, CB * CNQ, 4 * CF, CF, 1);
  launch_gemm<true, 1, false, true>(stream, h1, 0, f1_w2, 0, f1_b2, inter3, 0,
                                    out_inter, 0, CB * CNQ, CF, 4 * CF, 1);

  rowln_kernel<<<dim3((CB * CNC + 7) / 8), blk, 0, stream>>>(intra3, lnbuf, f2_g, f2_b, CB * CNC);
  launch_gemm<true, 1, true, false>(stream, lnbuf, 0, f2_w1, 0, f2_b1, nullptr, 0,
                                    h1, 0, CB * CNC, 4 * CF, CF, 1);
  launch_gemm<true, 1, false, true>(stream, h1, 0, f2_w2, 0, f2_b2, intra3, 0,
                                    out_intra, 0, CB * CNC, CF, 4 * CF, 1);
}